// MappingNetwork_42520176230546
// MI455X (gfx1250) — compile-verified
//
#include <hip/hip_runtime.h>

// Problem constants (match reference)
#define B_SAMP 16384
#define LATENT 16
#define HID    512
#define STYLE  64
#define NDOM   8
#define MTILE  64
#define PTOT   (B_SAMP + NDOM * MTILE)   // 16896 = 264 * 64, padded-compact row space

typedef __attribute__((ext_vector_type(2))) float v2f;
typedef __attribute__((ext_vector_type(8))) float v8f;

// ---------------- binning kernels ----------------

__global__ __launch_bounds__(256) void init_kernel(int* counts, int* perm) {
    int i = blockIdx.x * blockDim.x + threadIdx.x;
    if (i < NDOM) counts[i] = 0;
    if (i < PTOT) perm[i] = 0;   // padding rows read sample 0 (harmless, masked at output)
}

__global__ __launch_bounds__(256) void count_kernel(const int* __restrict__ labels, int* counts) {
    int i = blockIdx.x * blockDim.x + threadIdx.x;
    if (i < B_SAMP) atomicAdd(&counts[labels[i]], 1);
}

__global__ void scan_kernel(const int* __restrict__ counts, int* offsets, int* cursors) {
    if (threadIdx.x == 0) {
        int off = 0;
        for (int d = 0; d < NDOM; ++d) {
            offsets[d] = off;
            cursors[d] = off;
            off += (counts[d] + MTILE - 1) & ~(MTILE - 1);   // pad to M-tile
        }
        offsets[NDOM] = off;
    }
}

__global__ __launch_bounds__(256) void scatter_kernel(const int* __restrict__ labels,
                                                      int* cursors, int* perm) {
    int i = blockIdx.x * blockDim.x + threadIdx.x;
    if (i < B_SAMP) {
        int d = labels[i];
        int pos = atomicAdd(&cursors[d], 1);
        perm[pos] = i;
    }
}

// ---------------- WMMA GEMM ----------------
// C = act(A @ W + b), f32 end-to-end via V_WMMA_F32_16X16X4_F32.
// MODE: 0 = trunk (dense rows)
//       1 = branch entry (gather input rows via perm, per-domain weights)
//       2 = branch mid   (compact rows, per-domain weights)
//       3 = branch final (compact rows in, masked scatter to output, no ReLU)
// Wave tile: 64 x 64 (4x4 of 16x16 WMMA tiles, 16 v8f accumulators).
//   -> 12 loads (4x b64 A + 8x b32 B) per 16 WMMA per K-step.
// Block tile: 64 x (NWAVE*64); one wave-row so the 64-row domain padding holds.
// All loads: uniform SGPR base + 32-bit lane index + compile-time immediate.

template <int NWAVE, int MODE, int K, int N>
__global__ __launch_bounds__(NWAVE * 32)
void gemm_kernel(const float* __restrict__ A, const float* __restrict__ W,
                 const float* __restrict__ bias, float* __restrict__ Cout,
                 const int* __restrict__ perm, const int* __restrict__ offsets,
                 const int* __restrict__ counts)
{
    const int tid     = threadIdx.x;
    const int wn      = tid >> 5;           // wave column index
    const int lane    = tid & 31;
    const int lane_lo = lane & 15;
    const int lane_hi = lane >> 4;          // 0 for lanes 0-15, 1 for lanes 16-31

    const int p_base = blockIdx.y * MTILE;

    int d = 0;
    if (MODE != 0) {
        const int total = offsets[NDOM];
        if (p_base >= total) return;        // uniform early-exit: EXEC stays all-1s for WMMA
        #pragma unroll
        for (int i = 1; i < NDOM; ++i)
            if (p_base >= offsets[i]) d = i;
    }

    const int n0 = blockIdx.x * (NWAVE * 64) + wn * 64;

    // Per-lane 32-bit element indices (bases stay uniform in SGPRs).
    // A fragment (16x4): lanes 0-15 and 16-31 hold the same 16 rows; lane_hi
    // selects K pair {0,1} vs {2,3}.
    int aidx[4];
    #pragma unroll
    for (int mi = 0; mi < 4; ++mi) {
        const int p = p_base + mi * 16 + lane_lo;
        const int r = (MODE == 1) ? perm[p] : p;
        aidx[mi] = r * K + 2 * lane_hi;
    }
    // B fragment (4x16): lane_lo selects the N column, lane_hi selects K pair.
    // Fold the (uniform) per-domain weight offset into the 32-bit index.
    const int dwoff = d * K * N;
    int bidx[4];
    #pragma unroll
    for (int ni = 0; ni < 4; ++ni)
        bidx[ni] = dwoff + 2 * lane_hi * N + (n0 + ni * 16 + lane_lo);

    v8f acc[4][4] = {};

    #pragma unroll 8
    for (int k = 0; k < K; k += 4) {
        v2f a[4], b[4];
        #pragma unroll
        for (int mi = 0; mi < 4; ++mi)
            a[mi] = *(const v2f*)(A + (aidx[mi] + k));     // b64, imm offset k*4
        #pragma unroll
        for (int ni = 0; ni < 4; ++ni) {
            v2f t;
            t.x = W[bidx[ni] + k * N];                     // b32, imm offset k*N*4
            t.y = W[bidx[ni] + (k + 1) * N];               // b32, imm offset (k+1)*N*4
            b[ni] = t;
        }
        #pragma unroll
        for (int mi = 0; mi < 4; ++mi)
            #pragma unroll
            for (int ni = 0; ni < 4; ++ni)
                acc[mi][ni] = __builtin_amdgcn_wmma_f32_16x16x4_f32(
                    false, a[mi], false, b[ni], (short)0, acc[mi][ni], false, false);
    }

    // Epilogue: bias + optional ReLU + store.
    // C/D layout: VGPR r -> row (r + 8*lane_hi), col lane_lo within a 16x16 tile.
    float bv[4];
    #pragma unroll
    for (int ni = 0; ni < 4; ++ni)
        bv[ni] = bias[d * N + n0 + ni * 16 + lane_lo];

    #pragma unroll
    for (int mi = 0; mi < 4; ++mi) {
        #pragma unroll
        for (int ni = 0; ni < 4; ++ni) {
            const int colg = n0 + ni * 16 + lane_lo;
            #pragma unroll
            for (int r = 0; r < 8; ++r) {
                const int rowp = p_base + mi * 16 + r + 8 * lane_hi;
                float v = acc[mi][ni][r] + bv[ni];
                if (MODE != 3) v = fmaxf(v, 0.0f);         // ReLU on all but output layer
                if (MODE == 3) {
                    if (rowp - offsets[d] < counts[d]) {
                        const int s = perm[rowp];
                        Cout[s * N + colg] = v;            // scatter to d_out
                    }
                } else {
                    Cout[rowp * N + colg] = v;
                }
            }
        }
    }
}

// ---------------- host launcher ----------------

extern "C" void kernel_launch(void* const* d_in, const int* in_sizes, int n_in,
                              void* d_out, int out_size, void* d_ws, size_t ws_size,
                              hipStream_t stream) {
    (void)in_sizes; (void)n_in; (void)out_size; (void)ws_size;

    const float* z      = (const float*)d_in[0];
    const int*   labels = (const int*)  d_in[1];
    const float* sw0 = (const float*)d_in[2];  const float* sb0 = (const float*)d_in[3];
    const float* sw1 = (const float*)d_in[4];  const float* sb1 = (const float*)d_in[5];
    const float* sw2 = (const float*)d_in[6];  const float* sb2 = (const float*)d_in[7];
    const float* sw3 = (const float*)d_in[8];  const float* sb3 = (const float*)d_in[9];
    const float* uw1 = (const float*)d_in[10]; const float* ub1 = (const float*)d_in[11];
    const float* uw2 = (const float*)d_in[12]; const float* ub2 = (const float*)d_in[13];
    const float* uw3 = (const float*)d_in[14]; const float* ub3 = (const float*)d_in[15];
    const float* uwo = (const float*)d_in[16]; const float* ubo = (const float*)d_in[17];
    float* out = (float*)d_out;

    char* ws = (char*)d_ws;
    int* counts  = (int*)(ws);          // 8 ints
    int* cursors = (int*)(ws + 64);     // 8 ints
    int* offsets = (int*)(ws + 128);    // 9 ints
    int* perm    = (int*)(ws + 256);    // PTOT ints
    float* actA  = (float*)(ws + (1u << 20));
    float* actB  = actA + (size_t)PTOT * HID;   // two 34.6MB ping-pong buffers (L2-resident)

    // 1) bin samples by domain into a padded-compact permutation
    init_kernel   <<<(PTOT + 255) / 256, 256, 0, stream>>>(counts, perm);
    count_kernel  <<<(B_SAMP + 255) / 256, 256, 0, stream>>>(labels, counts);
    scan_kernel   <<<1, 32, 0, stream>>>(counts, offsets, cursors);
    scatter_kernel<<<(B_SAMP + 255) / 256, 256, 0, stream>>>(labels, cursors, perm);

    // 2) shared trunk: B x {16,512,512,512} -> 512, ReLU
    dim3 gT(HID / 256, B_SAMP / MTILE);   // 2 x 256, 128-thread blocks (4 waves, 64x256 tile)
    gemm_kernel<4, 0, LATENT, HID><<<gT, 128, 0, stream>>>(z,    sw0, sb0, actA, nullptr, nullptr, nullptr);
    gemm_kernel<4, 0, HID,    HID><<<gT, 128, 0, stream>>>(actA, sw1, sb1, actB, nullptr, nullptr, nullptr);
    gemm_kernel<4, 0, HID,    HID><<<gT, 128, 0, stream>>>(actB, sw2, sb2, actA, nullptr, nullptr, nullptr);
    gemm_kernel<4, 0, HID,    HID><<<gT, 128, 0, stream>>>(actA, sw3, sb3, actB, nullptr, nullptr, nullptr);

    // 3) per-domain branches on gathered rows only (8x less work than dense reference)
    dim3 gB(HID / 256, PTOT / MTILE);     // 2 x 264 (blocks past offsets[NDOM] exit early)
    gemm_kernel<4, 1, HID, HID><<<gB, 128, 0, stream>>>(actB, uw1, ub1, actA, perm, offsets, counts);
    gemm_kernel<4, 2, HID, HID><<<gB, 128, 0, stream>>>(actA, uw2, ub2, actB, perm, offsets, counts);
    gemm_kernel<4, 2, HID, HID><<<gB, 128, 0, stream>>>(actB, uw3, ub3, actA, perm, offsets, counts);

    // 4) output layer 512 -> 64, no ReLU, masked scatter to per-sample rows of d_out
    dim3 gF(1, PTOT / MTILE);             // 1 wave per block (64x64 tile)
    gemm_kernel<1, 3, HID, STYLE><<<gF, 32, 0, stream>>>(actA, uwo, ubo, out, perm, offsets, counts);
}